// GCN_11149735101024
// MI455X (gfx1250) — compile-verified
//
#include <hip/hip_runtime.h>

typedef __attribute__((ext_vector_type(2))) float v2f;
typedef __attribute__((ext_vector_type(8))) float v8f;

static constexpr int NN = 100000;   // nodes
static constexpr int NE = 1600000;  // edges
static constexpr int DF = 128;      // input features
static constexpr int DH = 64;       // hidden
static constexpr int DC = 40;       // classes
#define NEG_SLOPE 0.01f

__device__ __forceinline__ float lrelu(float x) { return x > 0.f ? x : x * NEG_SLOPE; }

// Exact f32 WMMA: D(16x16) = A(16x4) * B(4x16) + C
__device__ __forceinline__ v8f wmma4(v2f a, v2f b, v8f c) {
    return __builtin_amdgcn_wmma_f32_16x16x4_f32(false, a, false, b, (short)0, c, false, false);
}

__global__ void zero_f32(float* __restrict__ p, int n) {
    int i = blockIdx.x * blockDim.x + threadIdx.x;
    int stride = gridDim.x * blockDim.x;
    for (; i < n; i += stride) p[i] = 0.f;
}

__global__ void deg_kernel(const int* __restrict__ snd, const int* __restrict__ rcv,
                           float* __restrict__ sdeg, float* __restrict__ rdeg) {
    int e = blockIdx.x * blockDim.x + threadIdx.x;
    if (e < NE) {
        atomicAdd(&sdeg[snd[e]], 1.f);
        atomicAdd(&rdeg[rcv[e]], 1.f);
    }
}

// Fused: h1 = lrelu(lrelu(X@W1+b1)@W2+b2) * rsqrt(max(sender_deg,1))
// One wave handles a 16-row tile; 8 waves per block.
__launch_bounds__(256)
__global__ void mlp_kernel(const float* __restrict__ X,
                           const float* __restrict__ W1, const float* __restrict__ b1,
                           const float* __restrict__ W2, const float* __restrict__ b2,
                           const float* __restrict__ sdeg,
                           float* __restrict__ h1, int nTiles) {
    __shared__ float lds[8 * 16 * DH];  // 32 KB: per-wave C->A relayout scratch
    const int w    = threadIdx.x >> 5;
    const int lane = threadIdx.x & 31;
    const int m16  = lane & 15;   // A: row in tile / B,C,D: column in n-tile
    const int kh   = lane >> 4;   // lane half selects K pair (A/B) or M+8 (C/D)
    const int t    = blockIdx.x * 8 + w;
    const bool active = (t < nTiles);
    const int row0 = t * 16;
    float* my = &lds[w * 16 * DH];

    if (active) {
        // ---- Layer 1: X(16x128) @ W1(128x64) ----
        v8f c0 = {}, c1 = {}, c2 = {}, c3 = {};
        #pragma unroll 4
        for (int kb = 0; kb < DF; kb += 4) {
            v2f a = *(const v2f*)(X + (size_t)(row0 + m16) * DF + kb + 2 * kh);
            const float* wr0 = W1 + (size_t)(kb + 2 * kh) * DH;
            const float* wr1 = wr0 + DH;
            v2f b;
            b.x = wr0[m16];      b.y = wr1[m16];      c0 = wmma4(a, b, c0);
            b.x = wr0[16 + m16]; b.y = wr1[16 + m16]; c1 = wmma4(a, b, c1);
            b.x = wr0[32 + m16]; b.y = wr1[32 + m16]; c2 = wmma4(a, b, c2);
            b.x = wr0[48 + m16]; b.y = wr1[48 + m16]; c3 = wmma4(a, b, c3);
        }
        // bias + leaky-relu, relayout C(16x64) into LDS row-major for layer-2 A frags
        #pragma unroll
        for (int r = 0; r < 8; ++r) {
            float* dr = my + (r + 8 * kh) * DH;
            dr[m16]      = lrelu(c0[r] + b1[m16]);
            dr[16 + m16] = lrelu(c1[r] + b1[16 + m16]);
            dr[32 + m16] = lrelu(c2[r] + b1[32 + m16]);
            dr[48 + m16] = lrelu(c3[r] + b1[48 + m16]);
        }
    }
    __syncthreads();
    if (!active) return;

    // ---- Layer 2: h(16x64) @ W2(64x64) ----
    v8f d0 = {}, d1 = {}, d2 = {}, d3 = {};
    #pragma unroll 4
    for (int kb = 0; kb < DH; kb += 4) {
        v2f a = *(const v2f*)(my + m16 * DH + kb + 2 * kh);
        const float* wr0 = W2 + (size_t)(kb + 2 * kh) * DH;
        const float* wr1 = wr0 + DH;
        v2f b;
        b.x = wr0[m16];      b.y = wr1[m16];      d0 = wmma4(a, b, d0);
        b.x = wr0[16 + m16]; b.y = wr1[16 + m16]; d1 = wmma4(a, b, d1);
        b.x = wr0[32 + m16]; b.y = wr1[32 + m16]; d2 = wmma4(a, b, d2);
        b.x = wr0[48 + m16]; b.y = wr1[48 + m16]; d3 = wmma4(a, b, d3);
    }
    float sc[8];
    #pragma unroll
    for (int r = 0; r < 8; ++r) sc[r] = rsqrtf(fmaxf(sdeg[row0 + r + 8 * kh], 1.f));
    #pragma unroll
    for (int r = 0; r < 8; ++r) {
        float* orow = h1 + (size_t)(row0 + r + 8 * kh) * DH;
        orow[m16]      = lrelu(d0[r] + b2[m16])      * sc[r];
        orow[16 + m16] = lrelu(d1[r] + b2[16 + m16]) * sc[r];
        orow[32 + m16] = lrelu(d2[r] + b2[32 + m16]) * sc[r];
        orow[48 + m16] = lrelu(d3[r] + b2[48 + m16]) * sc[r];
    }
}

// Edge scatter for layer 1: agg[rcv] += h1[snd]  (64 features per edge)
__global__ void agg1_kernel(const float* __restrict__ h1, const int* __restrict__ snd,
                            const int* __restrict__ rcv, float* __restrict__ agg) {
    int idx = blockIdx.x * blockDim.x + threadIdx.x;
    int e = idx >> 6, f = idx & 63;
    if (e < NE)
        atomicAdd(&agg[(size_t)rcv[e] * DH + f], h1[(size_t)snd[e] * DH + f]);
}

// h2 = ((agg1 * rsqrt(rdeg)) @ W3 + b3) * rsqrt(sdeg)   (64 -> 40, padded to 48)
__launch_bounds__(256)
__global__ void fin1_kernel(const float* __restrict__ agg1,
                            const float* __restrict__ rdeg, const float* __restrict__ sdeg,
                            const float* __restrict__ W3, const float* __restrict__ b3,
                            float* __restrict__ h2, int nTiles) {
    const int w    = threadIdx.x >> 5;
    const int lane = threadIdx.x & 31;
    const int m16  = lane & 15;
    const int kh   = lane >> 4;
    const int t    = blockIdx.x * 8 + w;
    if (t >= nTiles) return;              // wave-uniform: EXEC all-1s inside
    const int row0 = t * 16;
    const float rs = rsqrtf(fmaxf(rdeg[row0 + m16], 1.f));  // per-A-row scale

    v8f c0 = {}, c1 = {}, c2 = {};
    #pragma unroll 4
    for (int kb = 0; kb < DH; kb += 4) {
        v2f a = *(const v2f*)(agg1 + (size_t)(row0 + m16) * DH + kb + 2 * kh);
        a.x *= rs; a.y *= rs;
        const float* wr0 = W3 + (size_t)(kb + 2 * kh) * DC;
        const float* wr1 = wr0 + DC;
        v2f b;
        b.x = wr0[m16];      b.y = wr1[m16];      c0 = wmma4(a, b, c0);
        b.x = wr0[16 + m16]; b.y = wr1[16 + m16]; c1 = wmma4(a, b, c1);
        const int n2 = 32 + m16;                       // pad columns 40..47 with 0
        b.x = (n2 < DC) ? wr0[n2] : 0.f;
        b.y = (n2 < DC) ? wr1[n2] : 0.f;
        c2 = wmma4(a, b, c2);
    }
    float sc[8];
    #pragma unroll
    for (int r = 0; r < 8; ++r) sc[r] = rsqrtf(fmaxf(sdeg[row0 + r + 8 * kh], 1.f));
    #pragma unroll
    for (int r = 0; r < 8; ++r) {
        float* orow = h2 + (size_t)(row0 + r + 8 * kh) * DC;
        orow[m16]      = (c0[r] + b3[m16])      * sc[r];
        orow[16 + m16] = (c1[r] + b3[16 + m16]) * sc[r];
        if (32 + m16 < DC)
            orow[32 + m16] = (c2[r] + b3[32 + m16]) * sc[r];
    }
}

// Edge scatter for layer 2: out[rcv] += h2[snd]  (40 features per edge)
__global__ void agg2_kernel(const float* __restrict__ h2, const int* __restrict__ snd,
                            const int* __restrict__ rcv, float* __restrict__ out) {
    int idx = blockIdx.x * blockDim.x + threadIdx.x;
    int e = idx / DC, f = idx - e * DC;
    if (e < NE)
        atomicAdd(&out[(size_t)rcv[e] * DC + f], h2[(size_t)snd[e] * DC + f]);
}

// out = softmax(out * rsqrt(max(rdeg,1)), axis=-1)
__global__ void softmax_kernel(float* __restrict__ out, const float* __restrict__ rdeg) {
    int n = blockIdx.x * blockDim.x + threadIdx.x;
    if (n >= NN) return;
    float rs = rsqrtf(fmaxf(rdeg[n], 1.f));
    float* row = out + (size_t)n * DC;
    float v[DC];
    float mx = -3.402823466e38f;
    #pragma unroll
    for (int f = 0; f < DC; ++f) { v[f] = row[f] * rs; mx = fmaxf(mx, v[f]); }
    float s = 0.f;
    #pragma unroll
    for (int f = 0; f < DC; ++f) { v[f] = __expf(v[f] - mx); s += v[f]; }
    float inv = 1.f / s;
    #pragma unroll
    for (int f = 0; f < DC; ++f) row[f] = v[f] * inv;
}

extern "C" void kernel_launch(void* const* d_in, const int* in_sizes, int n_in,
                              void* d_out, int out_size, void* d_ws, size_t ws_size,
                              hipStream_t stream) {
    const float* nodes     = (const float*)d_in[0];
    const int*   senders   = (const int*)d_in[1];
    const int*   receivers = (const int*)d_in[2];
    const float* W1 = (const float*)d_in[3];
    const float* b1 = (const float*)d_in[4];
    const float* W2 = (const float*)d_in[5];
    const float* b2 = (const float*)d_in[6];
    const float* W3 = (const float*)d_in[7];
    const float* b3 = (const float*)d_in[8];
    float* out = (float*)d_out;
    float* ws  = (float*)d_ws;

    // workspace layout (floats): sdeg[N] | rdeg[N] | h1[N*64] | agg1[N*64] | h2[N*40]
    float* sdeg = ws;
    float* rdeg = ws + NN;
    float* h1   = ws + 2 * (size_t)NN;
    float* agg1 = h1 + (size_t)NN * DH;
    float* h2   = agg1 + (size_t)NN * DH;

    const int nTiles = NN / 16;  // 6250, exact
    zero_f32<<<2048, 256, 0, stream>>>(sdeg, 2 * NN);
    zero_f32<<<8192, 256, 0, stream>>>(agg1, NN * DH);
    zero_f32<<<8192, 256, 0, stream>>>(out, NN * DC);
    deg_kernel<<<(NE + 255) / 256, 256, 0, stream>>>(senders, receivers, sdeg, rdeg);
    mlp_kernel<<<(nTiles + 7) / 8, 256, 0, stream>>>(nodes, W1, b1, W2, b2, sdeg, h1, nTiles);
    agg1_kernel<<<(NE * 64) / 256, 256, 0, stream>>>(h1, senders, receivers, agg1);
    fin1_kernel<<<(nTiles + 7) / 8, 256, 0, stream>>>(agg1, rdeg, sdeg, W3, b3, h2, nTiles);
    agg2_kernel<<<(NE * DC + 255) / 256, 256, 0, stream>>>(h2, senders, receivers, out);
    softmax_kernel<<<(NN + 255) / 256, 256, 0, stream>>>(out, rdeg);
}